// AdditiveAttention_41283225649206
// MI455X (gfx1250) — compile-verified
//
#include <hip/hip_runtime.h>
#include <math.h>

// Problem constants (match reference)
constexpr int kB = 8;
constexpr int kQ = 128;
constexpr int kV = 512;
constexpr int kH = 512;

typedef __attribute__((ext_vector_type(2))) float v2f;
typedef __attribute__((ext_vector_type(8))) float v8f;

// Hardware tanh on the TRANS pipe, guaranteed: builtin if present, else
// inline-asm v_tanh_f32 (gfx1250 VOP1 trans op). Never fall back to the
// multi-instruction ocml polynomial.
__device__ __forceinline__ float fast_tanh(float x) {
#if __has_builtin(__builtin_amdgcn_tanhf)
  return __builtin_amdgcn_tanhf(x);
#else
  float r;
  asm("v_tanh_f32 %0, %1" : "=v"(r) : "v"(x));
  return r;
#endif
}

// ---------------------------------------------------------------------------
// Generic fp32 WMMA GEMM: C[m,n] = sum_k A[m,k] * B(k,n) (+ bias[n])
//   BT=true : B(k,n) = Bp[n*ldb + k]  (weight matrices stored [N][K])
//   BT=false: B(k,n) = Bp[k*ldb + n]  (dense K-major operand, e.g. values)
//   CT=true : store C transposed: C[n*ldc + m]
// One wave computes one 16x16 tile via V_WMMA_F32_16X16X4_F32, K-step of 4.
// blockDim = (32, 4): 4 waves per block, each owning a distinct m-tile.
// ---------------------------------------------------------------------------
template <bool BT, bool CT>
__global__ void wmma_gemm_f32(const float* __restrict__ A, int lda,
                              const float* __restrict__ Bp, int ldb,
                              const float* __restrict__ bias,
                              float* __restrict__ C, int ldc, int K,
                              long long aStride, long long bStride,
                              long long cStride) {
  const int batch = blockIdx.z;
  const float* __restrict__ Ab = A + (long long)batch * aStride;
  const float* __restrict__ Bb = Bp + (long long)batch * bStride;
  float* __restrict__ Cb = C + (long long)batch * cStride;

  const int lane = threadIdx.x;       // 0..31
  const int lm = lane & 15;           // row/col within tile
  const int half = lane >> 4;         // 0 or 1
  const int n0 = blockIdx.x * 16;
  const int m0 = (blockIdx.y * blockDim.y + threadIdx.y) * 16;

  v8f acc = {};
#pragma unroll 2
  for (int k0 = 0; k0 < K; k0 += 4) {
    const int kb = k0 + 2 * half;  // this lane-half supplies K = kb, kb+1
    // A fragment: element (m = m0+lm, k = kb / kb+1), contiguous in k
    const float2 av =
        *reinterpret_cast<const float2*>(Ab + (long long)(m0 + lm) * lda + kb);
    v2f a;
    a[0] = av.x;
    a[1] = av.y;
    // B fragment: element (k = kb / kb+1, n = n0+lm)
    v2f bf;
    if (BT) {
      const float2 bvv = *reinterpret_cast<const float2*>(
          Bb + (long long)(n0 + lm) * ldb + kb);
      bf[0] = bvv.x;
      bf[1] = bvv.y;
    } else {
      bf[0] = Bb[(long long)kb * ldb + (n0 + lm)];
      bf[1] = Bb[(long long)(kb + 1) * ldb + (n0 + lm)];
    }
    acc = __builtin_amdgcn_wmma_f32_16x16x4_f32(
        /*neg_a=*/false, a, /*neg_b=*/false, bf,
        /*c_mod=*/(short)0, acc, /*reuse_a=*/false, /*reuse_b=*/false);
  }

  const float bsv = bias ? bias[n0 + lm] : 0.0f;
#pragma unroll
  for (int r = 0; r < 8; ++r) {
    const int m = m0 + r + 8 * half;  // C/D layout: VGPR r -> row r (+8 hi half)
    const float val = acc[r] + bsv;
    if (CT)
      Cb[(long long)(n0 + lm) * ldc + m] = val;
    else
      Cb[(long long)m * ldc + (n0 + lm)] = val;
  }
}

// ---------------------------------------------------------------------------
// Fused scores + mask + softmax.
// One block per (b, q); 128 threads, each owning 4 consecutive v columns:
//   score[v] = sum_h tanh(qp[b,q,h] + vpT[b,h,v]) * wc[h] + bc
// Per h: one coalesced global_load_b128 of vpT + two LDS broadcasts feed
// 4 x (v_add / v_tanh / v_fma). qp row and wc staged in LDS.
// ---------------------------------------------------------------------------
constexpr int kVPT = 4;             // v columns per thread
constexpr int kThr = kV / kVPT;     // 128 threads per block

__global__ void scores_softmax_kernel(const float* __restrict__ qp,
                                      const float* __restrict__ vpT,
                                      const int* __restrict__ mask,
                                      const float* __restrict__ wc,
                                      const float* __restrict__ bc,
                                      float* __restrict__ weights) {
  __shared__ float sq[kH];
  __shared__ float swc[kH];
  __shared__ float red[kThr];

  const int q = blockIdx.x;
  const int b = blockIdx.y;
  const int t = threadIdx.x;  // 0..127

  for (int h = t; h < kH; h += kThr) {
    sq[h] = qp[((size_t)b * kQ + q) * kH + h];
    swc[h] = wc[h];
  }
  __syncthreads();

  const int v0 = t * kVPT;
  const float* __restrict__ vbase = vpT + (size_t)b * kH * kV + v0;

  float a0 = 0.0f, a1 = 0.0f, a2 = 0.0f, a3 = 0.0f;
#pragma unroll 2
  for (int h = 0; h < kH; ++h) {
    const float4 vv = *reinterpret_cast<const float4*>(vbase + (size_t)h * kV);
    const float s = sq[h];
    const float w = swc[h];
    a0 = fmaf(fast_tanh(s + vv.x), w, a0);
    a1 = fmaf(fast_tanh(s + vv.y), w, a1);
    a2 = fmaf(fast_tanh(s + vv.z), w, a2);
    a3 = fmaf(fast_tanh(s + vv.w), w, a3);
  }

  const float bc0 = bc[0];
  float s0 = a0 + bc0, s1 = a1 + bc0, s2 = a2 + bc0, s3 = a3 + bc0;
  const int4 mk = *reinterpret_cast<const int4*>(mask + (size_t)b * kV + v0);
  const float ninf = -__builtin_inff();
  if (mk.x == 0) s0 = ninf;
  if (mk.y == 0) s1 = ninf;
  if (mk.z == 0) s2 = ninf;
  if (mk.w == 0) s3 = ninf;

  // block max (4-way register pre-reduce, then 128-entry LDS tree)
  red[t] = fmaxf(fmaxf(s0, s1), fmaxf(s2, s3));
  __syncthreads();
  for (int s = kThr / 2; s > 0; s >>= 1) {
    if (t < s) red[t] = fmaxf(red[t], red[t + s]);
    __syncthreads();
  }
  const float mx = red[0];
  __syncthreads();

  const float e0 = __expf(s0 - mx);  // exp(-inf) -> 0 for masked columns
  const float e1 = __expf(s1 - mx);
  const float e2 = __expf(s2 - mx);
  const float e3 = __expf(s3 - mx);

  red[t] = (e0 + e1) + (e2 + e3);
  __syncthreads();
  for (int s = kThr / 2; s > 0; s >>= 1) {
    if (t < s) red[t] += red[t + s];
    __syncthreads();
  }
  const float inv = 1.0f / red[0];

  float4 out;
  out.x = e0 * inv;
  out.y = e1 * inv;
  out.z = e2 * inv;
  out.w = e3 * inv;
  *reinterpret_cast<float4*>(weights + ((size_t)b * kQ + q) * kV + v0) = out;
}

// ---------------------------------------------------------------------------
// Launch
// ---------------------------------------------------------------------------
extern "C" void kernel_launch(void* const* d_in, const int* in_sizes, int n_in,
                              void* d_out, int out_size, void* d_ws,
                              size_t ws_size, hipStream_t stream) {
  const float* query = (const float*)d_in[0];   // [B,Q,H]
  const float* values = (const float*)d_in[1];  // [B,V,H]
  const int* mask = (const int*)d_in[2];        // [B,V]
  const float* Wq = (const float*)d_in[3];      // [H,H] stored [N][K]
  const float* bq = (const float*)d_in[4];      // [H]
  const float* Wv = (const float*)d_in[5];      // [H,H]
  const float* bvp = (const float*)d_in[6];     // [H]
  const float* wc = (const float*)d_in[7];      // [H]
  const float* bc = (const float*)d_in[8];      // [1]

  float* context = (float*)d_out;                   // [B,Q,H]
  float* weights = context + (size_t)kB * kQ * kH;  // [B,Q,V]

  float* qp = (float*)d_ws;                // [B*Q, H]
  float* vpT = qp + (size_t)kB * kQ * kH;  // [B, H, V]

  const dim3 blk(32, 4, 1);

  // qp = query . Wq^T + bq ; batch folded into M (B*Q rows, contiguous)
  wmma_gemm_f32<true, false>
      <<<dim3(kH / 16, (kB * kQ) / 64, 1), blk, 0, stream>>>(
          query, kH, Wq, kH, bq, qp, kH, kH, 0LL, 0LL, 0LL);

  // vpT[b][h][v] = (values[b] . Wv^T + bv)^T ; per-batch transposed store
  wmma_gemm_f32<true, true>
      <<<dim3(kH / 16, kV / 64, kB), blk, 0, stream>>>(
          values, kH, Wv, kH, bvp, vpT, kV, kH,
          (long long)kV * kH, 0LL, (long long)kH * kV);

  // scores + mask + softmax -> weights
  scores_softmax_kernel<<<dim3(kQ, kB, 1), dim3(kThr, 1, 1), 0, stream>>>(
      qp, vpT, mask, wc, bc, weights);

  // context[b] = weights[b] . values[b]  (B is K-major here)
  wmma_gemm_f32<false, false>
      <<<dim3(kH / 16, kQ / 64, kB), blk, 0, stream>>>(
          weights, kV, values, kH, nullptr, context, kH, kV,
          (long long)kQ * kV, (long long)kV * kH, (long long)kQ * kH);
}